// CustomActivation_31576599560968
// MI455X (gfx1250) — compile-verified
//
#include <hip/hip_runtime.h>

// out[i] = sum_k alpha[k, i%64] * relu(x[i] + gamma[k, i%64]),  i in [0, B*D)
// Memory-bound streaming kernel: b128 NT loads/stores + global_prefetch.
// Grid-stride in float4 units; stride*4 is a multiple of 64, so each thread's
// group base (i*4 & 63) is loop-invariant -> alpha/gamma live in registers.

typedef float v4f __attribute__((ext_vector_type(4)));  // native vector: NT-builtin legal

#define PF_DIST 2  // prefetch distance in grid strides

__global__ __launch_bounds__(256) void grouped_relu_mix_kernel(
    const float* __restrict__ x,
    const float* __restrict__ alpha,   // [3,64]
    const float* __restrict__ gamma,   // [3,64]
    float* __restrict__ out,
    long long n4,        // number of float4 elements
    long long n)         // total scalar elements (for tail)
{
    const long long t      = (long long)blockIdx.x * blockDim.x + threadIdx.x;
    const long long stride = (long long)gridDim.x * blockDim.x;

    // This thread's 4 consecutive d-slots within the 64-wide group pattern.
    // Invariant across the grid-stride loop (stride*4 % 64 == 0).
    const int gbase = (int)((t * 4) & 63);

    const v4f a0 = *reinterpret_cast<const v4f*>(alpha + 0 * 64 + gbase);
    const v4f a1 = *reinterpret_cast<const v4f*>(alpha + 1 * 64 + gbase);
    const v4f a2 = *reinterpret_cast<const v4f*>(alpha + 2 * 64 + gbase);
    const v4f g0 = *reinterpret_cast<const v4f*>(gamma + 0 * 64 + gbase);
    const v4f g1 = *reinterpret_cast<const v4f*>(gamma + 1 * 64 + gbase);
    const v4f g2 = *reinterpret_cast<const v4f*>(gamma + 2 * 64 + gbase);

    const v4f* __restrict__ x4 = reinterpret_cast<const v4f*>(x);
    v4f* __restrict__ o4       = reinterpret_cast<v4f*>(out);

    for (long long i = t; i < n4; i += stride) {
        // Keep HBM request queues full; speculative, silently dropped if OOB-translated.
        if (i + PF_DIST * stride < n4)
            __builtin_prefetch((const void*)(x4 + i + PF_DIST * stride), 0, 0);

        const v4f xv = __builtin_nontemporal_load(x4 + i);      // global_load_b128 NT
        v4f o;
        o.x = fmaf(a2.x, fmaxf(xv.x + g2.x, 0.0f),
              fmaf(a1.x, fmaxf(xv.x + g1.x, 0.0f),
                   a0.x * fmaxf(xv.x + g0.x, 0.0f)));
        o.y = fmaf(a2.y, fmaxf(xv.y + g2.y, 0.0f),
              fmaf(a1.y, fmaxf(xv.y + g1.y, 0.0f),
                   a0.y * fmaxf(xv.y + g0.y, 0.0f)));
        o.z = fmaf(a2.z, fmaxf(xv.z + g2.z, 0.0f),
              fmaf(a1.z, fmaxf(xv.z + g1.z, 0.0f),
                   a0.z * fmaxf(xv.z + g0.z, 0.0f)));
        o.w = fmaf(a2.w, fmaxf(xv.w + g2.w, 0.0f),
              fmaf(a1.w, fmaxf(xv.w + g1.w, 0.0f),
                   a0.w * fmaxf(xv.w + g0.w, 0.0f)));
        __builtin_nontemporal_store(o, o4 + i);                 // global_store_b128 NT
    }

    // Scalar tail (empty for the reference shapes: n % 4 == 0).
    for (long long j = n4 * 4 + t; j < n; j += stride) {
        const int gi = (int)(j & 63);
        const float xv = x[j];
        float acc =        alpha[gi]       * fmaxf(xv + gamma[gi],        0.0f);
        acc = fmaf(alpha[64 + gi],  fmaxf(xv + gamma[64 + gi],  0.0f), acc);
        acc = fmaf(alpha[128 + gi], fmaxf(xv + gamma[128 + gi], 0.0f), acc);
        out[j] = acc;
    }
}

extern "C" void kernel_launch(void* const* d_in, const int* in_sizes, int n_in,
                              void* d_out, int out_size, void* d_ws, size_t ws_size,
                              hipStream_t stream) {
    (void)n_in; (void)d_ws; (void)ws_size; (void)out_size;

    const float* x     = (const float*)d_in[0];   // [B, D] f32
    const float* alpha = (const float*)d_in[1];   // [3, 64] f32
    const float* gamma = (const float*)d_in[2];   // [3, 64] f32
    float* out         = (float*)d_out;           // [B, D] f32

    const long long n  = (long long)in_sizes[0];  // B*D = 33,554,432
    const long long n4 = n / 4;                   // 8,388,608 float4s

    const int block = 256;                        // 8 wave32 waves / block
    // ~8 float4 iterations per thread; cap block count for huge inputs.
    long long blocks = (n4 + (long long)block * 8 - 1) / ((long long)block * 8);
    if (blocks < 1)     blocks = 1;
    if (blocks > 65535) blocks = 65535;

    grouped_relu_mix_kernel<<<(int)blocks, block, 0, stream>>>(
        x, alpha, gamma, out, n4, n);
}